// TestAllReduceFusedAddRMSNormStaticQuantFP8Model_7095285973070
// MI455X (gfx1250) — compile-verified
//
#include <hip/hip_runtime.h>
#include <hip/hip_bf16.h>
#include <stdint.h>

// ---------------------------------------------------------------------------
// Fused AllReduce(identity) + Add + RMSNorm + static FP8 quant + dynamic
// per-tensor FP8 requant for MI455X (gfx1250).
//
// Memory-bound (~0.2 flop/byte, ~480 MiB traffic -> ~21us @ 23.3 TB/s).
// No WMMA by design (no matrix op in the workload). CDNA5 features used:
//   * global_load_async_to_lds_b128 + s_wait_asynccnt (weight staging)
//   * v_cvt_pk_fp8_f32 / v_cvt_pk_f32_fp8 hardware OCP-e4m3 converts
//   * non-temporal b128 loads/stores for the streaming 128MiB tensors
//   * wave32 shuffle (ds_bpermute) + LDS row reduction
//
// d_out layout (f32): [0, T*H)   = q_dyn as exact e4m3-representable floats
//                     [T*H, 2TH) = residual_out
// d_ws  layout:       [0..3]     = global max-abs bits (u32)
//                     [256 ...]  = T*H fp8 bytes (static-quant intermediate)
// ---------------------------------------------------------------------------

#define TPB            256
#define ROW_H          4096
#define F4_PER_ROW     (ROW_H / 4)            // 1024 float4 per row
#define F4_PER_THREAD  (F4_PER_ROW / TPB)     // 4 float4 per thread
#define FP8_MAX_F      448.0f
#define EPS_F          1e-6f

typedef __attribute__((ext_vector_type(4))) float f4;
typedef __attribute__((ext_vector_type(2))) float v2f;

#if __has_builtin(__builtin_amdgcn_cvt_pk_fp8_f32) && __has_builtin(__builtin_amdgcn_cvt_pk_f32_fp8)
#define USE_HW_FP8 1
#pragma message("CDNA5 probe: USE_HW_FP8=1 (v_cvt_pk_fp8_f32 path)")
#else
#define USE_HW_FP8 0
#pragma message("CDNA5 probe: USE_HW_FP8=0 (software e4m3 fallback)")
#endif

#if __has_builtin(__builtin_amdgcn_s_wait_asynccnt)
#pragma message("CDNA5 probe: s_wait_asynccnt builtin available")
#else
#pragma message("CDNA5 probe: s_wait_asynccnt via inline asm")
#endif

__device__ __forceinline__ float clamp448(float x) {
  return fminf(fmaxf(x, -FP8_MAX_F), FP8_MAX_F);
}

// ---------------- software e4m3fn fallback (RNE) ---------------------------
__device__ __forceinline__ unsigned int sw_enc1(float x) {
  unsigned int s = (__float_as_uint(x) >> 31) << 7;
  float ax = fabsf(x);
  if (ax < 0x1p-10f) return s;                 // rounds to +/-0
  int e; float m = frexpf(ax, &e);             // ax = m*2^e, m in [0.5,1)
  int p = e - 1;
  float q = (p < -6) ? 0x1p-9f : ldexpf(1.0f, p - 3);
  float r = rintf(ax / q) * q;                 // round to e4m3 grid (RNE)
  if (r < 0x1p-9f) return s;
  m = frexpf(r, &e); p = e - 1;
  unsigned int enc;
  if (p < -6) enc = (unsigned int)(r * 512.0f);                 // denormal
  else        enc = ((unsigned int)(p + 7) << 3) |
                    (unsigned int)rintf((m - 0.5f) * 16.0f);    // normal
  return enc | s;
}
__device__ __forceinline__ float sw_dec1(unsigned int b) {
  float sgn = (b & 0x80) ? -1.0f : 1.0f;
  unsigned int ef = (b >> 3) & 0xF, mant = b & 7;
  float v = (ef == 0) ? ldexpf((float)mant, -9)
                      : ldexpf(1.0f + (float)mant * 0.125f, (int)ef - 7);
  return sgn * v;
}

// pack 4 (already clamped) floats -> 4 e4m3 bytes in one dword
__device__ __forceinline__ unsigned int fp8x4_pack(float a, float b, float c, float d) {
#if USE_HW_FP8
  int pk = 0;
  pk = __builtin_amdgcn_cvt_pk_fp8_f32(a, b, pk, false);  // bytes 0,1
  pk = __builtin_amdgcn_cvt_pk_fp8_f32(c, d, pk, true);   // bytes 2,3
  return (unsigned int)pk;
#else
  return sw_enc1(a) | (sw_enc1(b) << 8) | (sw_enc1(c) << 16) | (sw_enc1(d) << 24);
#endif
}
__device__ __forceinline__ void fp8x4_unpack(unsigned int pk, float o[4]) {
#if USE_HW_FP8
  v2f lo = __builtin_amdgcn_cvt_pk_f32_fp8((int)pk, false);
  v2f hi = __builtin_amdgcn_cvt_pk_f32_fp8((int)pk, true);
  o[0] = lo.x; o[1] = lo.y; o[2] = hi.x; o[3] = hi.y;
#else
  o[0] = sw_dec1(pk & 0xFF);         o[1] = sw_dec1((pk >> 8) & 0xFF);
  o[2] = sw_dec1((pk >> 16) & 0xFF); o[3] = sw_dec1((pk >> 24) & 0xFF);
#endif
}

// ---------------- CDNA5 async global->LDS copy -----------------------------
__device__ __forceinline__ void async_gl2lds_b128(unsigned lds_byte_addr,
                                                  unsigned gl_byte_off,
                                                  const void* sbase) {
  // GVS mode: mem = SADDR(sgpr64) + VADDR(vgpr u32) ; dsaddr = VDST(vgpr u32)
  asm volatile("global_load_async_to_lds_b128 %0, %1, %2"
               :: "v"(lds_byte_addr), "v"(gl_byte_off), "s"(sbase)
               : "memory");
}
__device__ __forceinline__ void wait_async0() {
#if __has_builtin(__builtin_amdgcn_s_wait_asynccnt)
  __builtin_amdgcn_s_wait_asynccnt(0);
#else
  asm volatile("s_wait_asynccnt 0x0" ::: "memory");
#endif
}

// ---------------------------------------------------------------------------
__global__ void k_init_max(unsigned int* gmax) { *gmax = 0u; }

__global__ __launch_bounds__(TPB)
void k_fused_rmsnorm_quant(const float* __restrict__ hs,
                           const float* __restrict__ res,
                           const float* __restrict__ weight,
                           const float* __restrict__ scale,
                           float* __restrict__ residual_out,
                           unsigned int* __restrict__ q_ws,
                           unsigned int* __restrict__ gmax_bits) {
  __shared__ __align__(16) float w_lds[ROW_H];
  __shared__ float red[TPB / 32];

  const int tid = threadIdx.x;
  const int row = blockIdx.x;

  // --- async-stage weight (16 KiB) into LDS: 4 x b128 per thread ----------
  #pragma unroll
  for (int j = 0; j < F4_PER_THREAD; ++j) {
    const int c = tid + j * TPB;                           // float4 chunk id
    const unsigned laddr = (unsigned)(uintptr_t)&w_lds[c * 4];
    async_gl2lds_b128(laddr, (unsigned)(c * 16), (const void*)weight);
  }
  wait_async0();
  __syncthreads();

  // --- pass over the row: add + residual_out + sum of squares -------------
  const f4* __restrict__ h4 = (const f4*)hs  + (size_t)row * F4_PER_ROW;
  const f4* __restrict__ r4 = (const f4*)res + (size_t)row * F4_PER_ROW;
  f4*       __restrict__ o4 = (f4*)residual_out + (size_t)row * F4_PER_ROW;

  f4 a[F4_PER_THREAD];
  float ss = 0.0f;
  #pragma unroll
  for (int i = 0; i < F4_PER_THREAD; ++i) {
    const int idx = tid + i * TPB;
    const f4 h = __builtin_nontemporal_load(&h4[idx]);    // streamed once
    const f4 r = __builtin_nontemporal_load(&r4[idx]);
    const f4 v = h + r;
    __builtin_nontemporal_store(v, &o4[idx]);             // never re-read
    a[i] = v;
    ss = fmaf(v.x, v.x, fmaf(v.y, v.y, fmaf(v.z, v.z, fmaf(v.w, v.w, ss))));
  }

  // wave32 reduce, then cross-wave via LDS
  #pragma unroll
  for (int off = 16; off > 0; off >>= 1) ss += __shfl_xor(ss, off, 32);
  if ((tid & 31) == 0) red[tid >> 5] = ss;
  __syncthreads();
  float total = 0.0f;
  #pragma unroll
  for (int i = 0; i < TPB / 32; ++i) total += red[i];
  const float rstd  = rsqrtf(total * (1.0f / (float)ROW_H) + EPS_F);
  const float inv_s = 1.0f / scale[0];

  // --- normalize, static fp8 quant, dequant, track max |deq| --------------
  unsigned int* __restrict__ qrow = q_ws + (size_t)row * F4_PER_ROW;
  float lmax = 0.0f;
  #pragma unroll
  for (int i = 0; i < F4_PER_THREAD; ++i) {
    const int idx = tid + i * TPB;
    const f4 v = a[i];
    const f4 w = *(const f4*)&w_lds[idx * 4];
    const float n0 = clamp448(v.x * rstd * w.x * inv_s);
    const float n1 = clamp448(v.y * rstd * w.y * inv_s);
    const float n2 = clamp448(v.z * rstd * w.z * inv_s);
    const float n3 = clamp448(v.w * rstd * w.w * inv_s);
    const unsigned int pk = fp8x4_pack(n0, n1, n2, n3);
    float d[4];
    fp8x4_unpack(pk, d);
    lmax = fmaxf(lmax, fmaxf(fmaxf(fabsf(d[0]), fabsf(d[1])),
                             fmaxf(fabsf(d[2]), fabsf(d[3]))));
    qrow[idx] = pk;                                       // fp8 bytes, L2-hot
  }

  #pragma unroll
  for (int off = 16; off > 0; off >>= 1) lmax = fmaxf(lmax, __shfl_xor(lmax, off, 32));
  __syncthreads();                        // prior reads of red[] are done
  if ((tid & 31) == 0) red[tid >> 5] = lmax;
  __syncthreads();
  if (tid == 0) {
    float m = red[0];
    #pragma unroll
    for (int i = 1; i < TPB / 32; ++i) m = fmaxf(m, red[i]);
    atomicMax(gmax_bits, __float_as_uint(m));  // m >= 0: uint order == float order
  }
}

// one uint4 (16 fp8 bytes) per thread: single b128 load, 4x b128 nt stores
__global__ __launch_bounds__(TPB)
void k_dynamic_requant(const uint4* __restrict__ q_ws,
                       const unsigned int* __restrict__ gmax_bits,
                       float* __restrict__ q_out) {
  const size_t i = (size_t)blockIdx.x * TPB + threadIdx.x;
  const float gmax  = __uint_as_float(*gmax_bits);
  const float inv_d = FP8_MAX_F / fmaxf(gmax, 1e-12f);     // 1/dyn_scale

  const uint4 pk4 = q_ws[i];                               // L2-resident
  const unsigned int pks[4] = {pk4.x, pk4.y, pk4.z, pk4.w};
  f4* __restrict__ out4 = (f4*)q_out + i * 4;

  #pragma unroll
  for (int k = 0; k < 4; ++k) {
    float d[4];
    fp8x4_unpack(pks[k], d);                               // deq (static)
    const float q0 = clamp448(d[0] * inv_d);
    const float q1 = clamp448(d[1] * inv_d);
    const float q2 = clamp448(d[2] * inv_d);
    const float q3 = clamp448(d[3] * inv_d);
    const unsigned int pk = fp8x4_pack(q0, q1, q2, q3);    // dynamic quant
    float e[4];
    fp8x4_unpack(pk, e);                                   // exact e4m3 values
    f4 o; o.x = e[0]; o.y = e[1]; o.z = e[2]; o.w = e[3];
    __builtin_nontemporal_store(o, &out4[k]);
  }
}

// ---------------------------------------------------------------------------
extern "C" void kernel_launch(void* const* d_in, const int* in_sizes, int n_in,
                              void* d_out, int out_size, void* d_ws, size_t ws_size,
                              hipStream_t stream) {
  (void)n_in; (void)out_size; (void)ws_size;
  const float* hs  = (const float*)d_in[0];   // [T,H] f32
  const float* res = (const float*)d_in[1];   // [T,H] f32
  const float* w   = (const float*)d_in[2];   // [H]   f32
  const float* sc  = (const float*)d_in[3];   // [1]   f32

  const int rows = in_sizes[0] / ROW_H;       // T = 8192

  float* out       = (float*)d_out;
  float* q_out     = out;                                   // T*H floats
  float* resid_out = out + (size_t)rows * ROW_H;            // T*H floats

  unsigned int* gmax = (unsigned int*)d_ws;
  unsigned int* q_ws = (unsigned int*)((char*)d_ws + 256);  // T*H fp8 bytes

  k_init_max<<<dim3(1), dim3(1), 0, stream>>>(gmax);
  k_fused_rmsnorm_quant<<<dim3(rows), dim3(TPB), 0, stream>>>(
      hs, res, w, sc, resid_out, q_ws, gmax);

  // 16 fp8 per thread
  const unsigned blocks16 = (unsigned)(((size_t)rows * ROW_H / 16) / TPB);
  k_dynamic_requant<<<dim3(blocks16), dim3(TPB), 0, stream>>>(
      (const uint4*)q_ws, gmax, q_out);
}